// RMAC_25297357374034
// MI455X (gfx1250) — compile-verified
//
#include <hip/hip_runtime.h>
#include <math.h>

typedef __attribute__((ext_vector_type(2))) float v2f;
typedef __attribute__((ext_vector_type(8))) float v8f;

static constexpr int kHW    = 196;   // 14*14 spatial elements per plane
static constexpr int kWdim  = 14;
static constexpr int kNReg  = 14;    // 1 full + 4 (9x9) + 9 (7x7); full counted x2 at combine
static constexpr float kEps = 1e-6f;

// ---------------------------------------------------------------------------
// Kernel 1: per-(b,c) plane, compute the 14 distinct region maxima in one read
// of x. Regions (H=W=14): full 14x14; 9x9 at row/col offsets {0,5}; 7x7 at
// row/col offsets {0,3,7}.
// ---------------------------------------------------------------------------
__global__ __launch_bounds__(256) void rmac_region_max_kernel(
    const float* __restrict__ x, float* __restrict__ m, int nPlanes) {
  int idx = blockIdx.x * blockDim.x + threadIdx.x;
  if (idx >= nPlanes) return;
  const float* p = x + (size_t)idx * kHW;

  float acc[kNReg];
#pragma unroll
  for (int r = 0; r < kNReg; ++r) acc[r] = -INFINITY;

#pragma unroll
  for (int i = 0; i < 14; ++i) {
    float row[14];
    const float2* p2 = (const float2*)(p + i * kWdim);  // rows are 8B-aligned
#pragma unroll
    for (int j = 0; j < 7; ++j) {
      float2 v = p2[j];
      row[2 * j] = v.x;
      row[2 * j + 1] = v.y;
    }
    // column-segment maxima
    float c3a = row[0];                                   // cols 0..6
#pragma unroll
    for (int j = 1; j <= 6; ++j) c3a = fmaxf(c3a, row[j]);
    float c3b = row[3];                                   // cols 3..9
#pragma unroll
    for (int j = 4; j <= 9; ++j) c3b = fmaxf(c3b, row[j]);
    float c3c = row[7];                                   // cols 7..13
#pragma unroll
    for (int j = 8; j <= 13; ++j) c3c = fmaxf(c3c, row[j]);
    float c2a  = fmaxf(c3a, fmaxf(row[7], row[8]));       // cols 0..8
    float c2b  = fmaxf(c3c, fmaxf(row[5], row[6]));       // cols 5..13
    float full = fmaxf(c2a, c3c);                         // cols 0..13

    // fold into region accumulators (compile-time row predicates)
    acc[0] = fmaxf(acc[0], full);
    if (i <= 8) { acc[1] = fmaxf(acc[1], c2a); acc[2] = fmaxf(acc[2], c2b); }
    if (i >= 5) { acc[3] = fmaxf(acc[3], c2a); acc[4] = fmaxf(acc[4], c2b); }
    if (i <= 6) {
      acc[5] = fmaxf(acc[5], c3a); acc[6] = fmaxf(acc[6], c3b); acc[7] = fmaxf(acc[7], c3c);
    }
    if (i >= 3 && i <= 9) {
      acc[8] = fmaxf(acc[8], c3a); acc[9] = fmaxf(acc[9], c3b); acc[10] = fmaxf(acc[10], c3c);
    }
    if (i >= 7) {
      acc[11] = fmaxf(acc[11], c3a); acc[12] = fmaxf(acc[12], c3b); acc[13] = fmaxf(acc[13], c3c);
    }
  }
#pragma unroll
  for (int r = 0; r < kNReg; ++r) m[(size_t)r * nPlanes + idx] = acc[r];
}

// ---------------------------------------------------------------------------
// Kernel 2: per (region, batch) L2 norm over C channels using
// V_WMMA_F32_16X16X4_F32. A 64-channel chunk is a 16x4 f32 A-tile; because
// the 4x16 B-tile layout of A^T is register-identical to A's layout, passing
// the same operand twice accumulates the Gram matrix; its diagonal trace is
// sum(m^2). One wave (32 lanes) per (r,b), EXEC all-ones throughout.
// ---------------------------------------------------------------------------
__global__ __launch_bounds__(32) void rmac_norm_kernel(
    const float* __restrict__ m, float* __restrict__ w, int C) {
  int rb = blockIdx.x;                 // r * B + b
  int lane = threadIdx.x;              // 0..31
  const float* base = m + (size_t)rb * C;

  // A-matrix layout 16x4 f32: lane L holds A[L&15][K0], A[L&15][K0+1],
  // K0 = (L<16 ? 0 : 2). Channel = (L&15)*4 + K0 + {0,1}.
  int off = (lane & 15) * 4 + ((lane >> 4) << 1);

  v8f acc = {0.f, 0.f, 0.f, 0.f, 0.f, 0.f, 0.f, 0.f};
  for (int k = 0; k < C; k += 64) {
    float2 t = *(const float2*)(base + k + off);
    v2f a;
    a.x = t.x;
    a.y = t.y;
    acc = __builtin_amdgcn_wmma_f32_16x16x4_f32(
        /*neg_a=*/false, a, /*neg_b=*/false, a,
        /*c_mod=*/(short)0, acc, /*reuse_a=*/false, /*reuse_b=*/false);
  }

  // Extract diagonal D[N][N]: lanes 0..7 -> acc[lane]; lanes 24..31 -> acc[lane-24].
  float d = 0.f;
  if (lane < 8)       d = acc[lane];
  else if (lane >= 24) d = acc[lane - 24];

#pragma unroll
  for (int s = 16; s >= 1; s >>= 1) d += __shfl_xor(d, s, 32);

  if (lane == 0) w[rb] = 1.0f / (sqrtf(d) + kEps);
}

// ---------------------------------------------------------------------------
// Kernel 3: out[b,c] = sum_r weight_r * m[r,b,c] * w[r,b]; full-image region
// weighted 2x (global MAC + identical level-1 region).
// ---------------------------------------------------------------------------
__global__ __launch_bounds__(256) void rmac_combine_kernel(
    const float* __restrict__ m, const float* __restrict__ w,
    float* __restrict__ out, int nPlanes, int C) {
  int idx = blockIdx.x * blockDim.x + threadIdx.x;
  if (idx >= nPlanes) return;
  int B = nPlanes / C;
  int b = idx / C;
  float v = 0.f;
#pragma unroll
  for (int r = 0; r < kNReg; ++r) {
    float wt = w[r * B + b];
    if (r == 0) wt *= 2.0f;
    v += m[(size_t)r * nPlanes + idx] * wt;
  }
  out[idx] = v;
}

extern "C" void kernel_launch(void* const* d_in, const int* in_sizes, int n_in,
                              void* d_out, int out_size, void* d_ws, size_t ws_size,
                              hipStream_t stream) {
  const float* x = (const float*)d_in[0];
  const int C = 2048;
  int nPlanes = in_sizes[0] / kHW;     // B*C = 131072
  int B = nPlanes / C;                 // 64

  float* m = (float*)d_ws;                      // [14][B][C] = 7.34 MB
  float* w = m + (size_t)kNReg * nPlanes;       // [14][B]    = 3.5 KB

  const int threads = 256;
  int blocks = (nPlanes + threads - 1) / threads;
  rmac_region_max_kernel<<<blocks, threads, 0, stream>>>(x, m, nPlanes);
  rmac_norm_kernel<<<kNReg * B, 32, 0, stream>>>(m, w, C);
  rmac_combine_kernel<<<blocks, threads, 0, stream>>>(m, w, (float*)d_out, nPlanes, C);
}